// SymtoGeo_15762529976766
// MI455X (gfx1250) — compile-verified
//
#include <hip/hip_runtime.h>
#include <math.h>

// ---------------- problem constants (match reference) ----------------
static const int HD  = 128;
static const int PD  = 1024;
static const int LD  = 1024;
static const int CD  = 128;
static const int NT  = 64, NH = 32, NA = 64, NB = 64, NPA = 32, NPE = 32;
static const int RT  = PD + LD + PD + LD + CD + NH;   // 4256 (geo_t rows)
static const int RH  = PD + PD + LD + CD;             // 3200 (geo_h rows)
static const int HH  = HD * HD;

// ---------------- CDNA5 WMMA (fp32, 16x16x4) ----------------
typedef float v2f __attribute__((ext_vector_type(2)));
typedef float v8f __attribute__((ext_vector_type(8)));

__device__ __forceinline__ v8f wmma_f32(v2f a, v2f b, v8f c) {
  // D = A(16x4,f32) * B(4x16,f32) + C(16x16,f32)  -> v_wmma_f32_16x16x4_f32
  return __builtin_amdgcn_wmma_f32_16x16x4_f32(false, a, false, b, (short)0, c,
                                               false, false);
}

// ---------------- mask reductions ----------------
// lpl_m[p] = (sum_l pl[p][l] >= 2); plp_m[l] = colsum(pl) >= 2; pcp_m[c] = colsum(pc) >= 2
__global__ void masks_kernel(const int* __restrict__ pl, const int* __restrict__ pc,
                             float* __restrict__ lpl_m, float* __restrict__ plp_m,
                             float* __restrict__ pcp_m) {
  int t = blockIdx.x * blockDim.x + threadIdx.x;
  if (t < PD) {
    int s = 0;
    for (int j = 0; j < LD; ++j) s += pl[(size_t)t * LD + j];
    lpl_m[t] = (s >= 2) ? 1.f : 0.f;
  }
  if (t < LD) {
    int s = 0;
    for (int i = 0; i < PD; ++i) s += pl[(size_t)i * LD + t];
    plp_m[t] = (s >= 2) ? 1.f : 0.f;
  }
  if (t < CD) {
    int s = 0;
    for (int i = 0; i < PD; ++i) s += pc[(size_t)i * CD + t];
    pcp_m[t] = (s >= 2) ? 1.f : 0.f;
  }
}

// ---------------- agg = relA @ margin  (relA: MxK of 0/1 int, margin: Kx128) ----------
// relA[m][k] = transA ? rel[k*ldA + m] : rel[m*ldA + k]
// One workgroup (8 wave32) computes a 16x128 output strip; A staged via LDS.
__global__ __launch_bounds__(256)
void agg_gemm_kernel(const int* __restrict__ rel, int ldA, int transA,
                     const float* __restrict__ Bm, int K, float* __restrict__ Cout) {
  __shared__ float sA[16 * 128];
  const int tid  = threadIdx.x;
  const int lane = tid & 31;
  const int w    = tid >> 5;                 // wave id 0..7 -> N tile
  const int m0   = blockIdx.x * 16;
  const int n    = w * 16 + (lane & 15);
  const int am   = lane & 15;
  const int koff = (lane >> 4) << 1;

  v8f acc = {0.f, 0.f, 0.f, 0.f, 0.f, 0.f, 0.f, 0.f};

  for (int k0 = 0; k0 < K; k0 += 128) {
    if (transA) {
      for (int i = tid; i < 16 * 128; i += 256) {
        int m = i & 15, k = i >> 4;
        sA[m * 128 + k] = (float)rel[(size_t)(k0 + k) * ldA + (m0 + m)];
      }
    } else {
      for (int i = tid; i < 16 * 128; i += 256) {
        int m = i >> 7, k = i & 127;
        sA[m * 128 + k] = (float)rel[(size_t)(m0 + m) * ldA + (k0 + k)];
      }
    }
    __syncthreads();
    for (int kk = 0; kk < 128; kk += 4) {
      v2f a = *(const v2f*)&sA[am * 128 + kk + koff];
      int kg = k0 + kk + koff;
      v2f b;
      b.x = Bm[(size_t)kg * HD + n];
      b.y = Bm[(size_t)(kg + 1) * HD + n];
      acc = wmma_f32(a, b, acc);
    }
    __syncthreads();
  }
  const int mb = (lane >> 4) << 3;
#pragma unroll
  for (int v = 0; v < 8; ++v)
    Cout[(size_t)(m0 + mb + v) * HD + n] = acc[v];
}

// ---------------- Y = [LN(ReLU(] (alpha*X0 + X1) @ W^T + b [))] ------------------
// W row-major [out=128][in=128] (PyTorch Linear). do_ln=0 -> raw GEMM (no bias).
__global__ __launch_bounds__(256)
void lrl_gemm_kernel(const float* __restrict__ X0, const float* __restrict__ X1,
                     float alpha, const float* __restrict__ W,
                     const float* __restrict__ bias, const float* __restrict__ g,
                     const float* __restrict__ be, float* __restrict__ Y, int do_ln) {
  __shared__ float sA[16 * 128];
  __shared__ float sC[16 * 128];
  __shared__ float sMean[16], sRstd[16];
  const int tid  = threadIdx.x;
  const int lane = tid & 31;
  const int w    = tid >> 5;
  const int m0   = blockIdx.x * 16;
  const int n    = w * 16 + (lane & 15);
  const int am   = lane & 15;
  const int koff = (lane >> 4) << 1;

  for (int i = tid; i < 16 * 128; i += 256) {
    int m = i >> 7, k = i & 127;
    float v = alpha * X0[(size_t)(m0 + m) * 128 + k];
    if (X1) v += X1[(size_t)(m0 + m) * 128 + k];
    sA[i] = v;
  }
  __syncthreads();

  v8f acc = {0.f, 0.f, 0.f, 0.f, 0.f, 0.f, 0.f, 0.f};
#pragma unroll 4
  for (int kk = 0; kk < 128; kk += 4) {
    v2f a = *(const v2f*)&sA[am * 128 + kk + koff];
    v2f b = *(const v2f*)&W[(size_t)n * 128 + kk + koff];  // B[k][n] = W[n][k]
    acc = wmma_f32(a, b, acc);
  }

  const int mb = (lane >> 4) << 3;
  if (!do_ln) {
#pragma unroll
    for (int v = 0; v < 8; ++v)
      Y[(size_t)(m0 + mb + v) * 128 + n] = acc[v];
    return;
  }
  float bn = bias[n];
#pragma unroll
  for (int v = 0; v < 8; ++v) {
    float x = acc[v] + bn;
    sC[(mb + v) * 128 + n] = x > 0.f ? x : 0.f;
  }
  __syncthreads();
  if (tid < 16) {
    float s = 0.f, s2 = 0.f;
    for (int k = 0; k < 128; ++k) {
      float x = sC[tid * 128 + k];
      s += x; s2 += x * x;
    }
    float mean = s * (1.f / 128.f);
    float var  = s2 * (1.f / 128.f) - mean * mean;
    sMean[tid] = mean;
    sRstd[tid] = rsqrtf(var + 1e-5f);
  }
  __syncthreads();
  for (int i = tid; i < 16 * 128; i += 256) {
    int m = i >> 7, k = i & 127;
    Y[(size_t)(m0 + m) * 128 + k] = (sC[i] - sMean[m]) * sRstd[m] * g[k] + be[k];
  }
}

// ---------------- concat masks ----------------
__global__ void mask_cat_kernel(const float* __restrict__ lpl_m,
                                const float* __restrict__ plp_m,
                                const float* __restrict__ pcp_m,
                                float* __restrict__ mask_t, float* __restrict__ mask_h) {
  int t = blockIdx.x * blockDim.x + threadIdx.x;
  if (t < RT) {
    float v;
    if (t < 2048)      v = 1.f;
    else if (t < 3072) v = lpl_m[t - 2048];
    else if (t < 4096) v = plp_m[t - 3072];
    else if (t < 4224) v = pcp_m[t - 4096];
    else               v = 1.f;
    mask_t[t] = v;
  }
  if (t < RH) {
    float v;
    if (t < 1024)      v = 1.f;
    else if (t < 2048) v = lpl_m[t - 1024];
    else if (t < 3072) v = plp_m[t - 2048];
    else               v = pcp_m[t - 3072];
    mask_h[t] = v;
  }
}

// ---------------- pair scoring: one wave32 per (sym,geo) pair ----------------
// h = ReLU(symW[i] + geoW[r] + b1); LN(h); logit = dot(h,W2)+b2; sigmoid(masked)
__global__ __launch_bounds__(256)
void pair_kernel(const float* __restrict__ symW, const float* __restrict__ geoW,
                 const float* __restrict__ mask, const float* __restrict__ b1,
                 const float* __restrict__ g, const float* __restrict__ be,
                 const float* __restrict__ W2, const float* __restrict__ b2p,
                 float* __restrict__ out, int Nsym, int R) {
  int gw = (int)(((unsigned)blockIdx.x * 256u + threadIdx.x) >> 5);
  if (gw >= Nsym * R) return;                       // uniform per wave
  int lane = threadIdx.x & 31;
  int i = gw / R;
  int r = gw - i * R;

  float4 s4 = ((const float4*)(symW + (size_t)i * 128))[lane];
  float4 g4 = ((const float4*)(geoW + (size_t)r * 128))[lane];
  float4 b4 = ((const float4*)b1)[lane];

  float x0 = s4.x + g4.x + b4.x; x0 = x0 > 0.f ? x0 : 0.f;
  float x1 = s4.y + g4.y + b4.y; x1 = x1 > 0.f ? x1 : 0.f;
  float x2 = s4.z + g4.z + b4.z; x2 = x2 > 0.f ? x2 : 0.f;
  float x3 = s4.w + g4.w + b4.w; x3 = x3 > 0.f ? x3 : 0.f;

  float s  = x0 + x1 + x2 + x3;
  float s2 = x0 * x0 + x1 * x1 + x2 * x2 + x3 * x3;
  for (int o = 16; o > 0; o >>= 1) {
    s  += __shfl_xor(s, o, 32);
    s2 += __shfl_xor(s2, o, 32);
  }
  float mean = s * (1.f / 128.f);
  float var  = s2 * (1.f / 128.f) - mean * mean;
  float rstd = rsqrtf(var + 1e-5f);

  float4 gg = ((const float4*)g)[lane];
  float4 bb = ((const float4*)be)[lane];
  float4 w2 = ((const float4*)W2)[lane];
  float d = ((x0 - mean) * rstd * gg.x + bb.x) * w2.x +
            ((x1 - mean) * rstd * gg.y + bb.y) * w2.y +
            ((x2 - mean) * rstd * gg.z + bb.z) * w2.z +
            ((x3 - mean) * rstd * gg.w + bb.w) * w2.w;
  for (int o = 16; o > 0; o >>= 1) d += __shfl_xor(d, o, 32);

  if (lane == 0) {
    float logit = d + b2p[0];
    float mk = mask ? mask[r] : 1.f;
    float z  = (mk > 0.f) ? logit : -1e9f;
    out[(size_t)i * R + r] = 1.f / (1.f + expf(-z));
  }
}

// ---------------- host orchestration ----------------
extern "C" void kernel_launch(void* const* d_in, const int* in_sizes, int n_in,
                              void* d_out, int out_size, void* d_ws, size_t ws_size,
                              hipStream_t stream) {
  (void)in_sizes; (void)n_in; (void)out_size; (void)ws_size;
  const float* points  = (const float*)d_in[0];
  const float* lines   = (const float*)d_in[1];
  const float* circles = (const float*)d_in[2];
  const float* text_s  = (const float*)d_in[3];
  const float* head_s  = (const float*)d_in[4];
  const float* angle_s = (const float*)d_in[5];
  const float* bar_s   = (const float*)d_in[6];
  const float* para_s  = (const float*)d_in[7];
  const float* perp_s  = (const float*)d_in[8];
  const float* rel_W   = (const float*)d_in[9];
  const float* rel_b   = (const float*)d_in[10];
  const float* rel_g   = (const float*)d_in[11];
  const float* rel_be  = (const float*)d_in[12];
  const float* gm_W    = (const float*)d_in[13];
  const float* gm_b    = (const float*)d_in[14];
  const float* gm_g    = (const float*)d_in[15];
  const float* gm_be   = (const float*)d_in[16];
  const float* sc_W1   = (const float*)d_in[17];
  const float* sc_b1   = (const float*)d_in[18];
  const float* sc_g    = (const float*)d_in[19];
  const float* sc_be   = (const float*)d_in[20];
  const float* sc_W2   = (const float*)d_in[21];
  const float* sc_b2   = (const float*)d_in[22];
  const int*   pl      = (const int*)d_in[23];
  const int*   pc      = (const int*)d_in[24];
  float* out = (float*)d_out;

  // workspace carve-out (~9.4 MB of floats)
  float* wsp = (float*)d_ws;
  size_t off = 0;
  auto alloc = [&](size_t n) { float* p = wsp + off; off += (n + 31) & ~(size_t)31; return p; };
  float* LPLb   = alloc((size_t)PD * HD);
  float* PLPb   = alloc((size_t)LD * HD);
  float* PCPb   = alloc((size_t)CD * HD);
  float* lpl_m  = alloc(PD);
  float* plp_m  = alloc(LD);
  float* pcp_m  = alloc(CD);
  float* mask_t = alloc(RT);
  float* mask_h = alloc(RH);
  float* geo_t  = alloc((size_t)RT * HD);
  float* geo_h  = alloc((size_t)RH * HD);
  float* tmp    = alloc((size_t)RT * HD);
  float* symWb  = alloc((size_t)64 * HD);
  float* geoWb  = alloc((size_t)RT * HD);

  // 1) relation masks
  masks_kernel<<<4, 256, 0, stream>>>(pl, pc, lpl_m, plp_m, pcp_m);

  // 2) LPL = LN(ReLU((2*pl@lines + points) @ relW0^T + b)) etc.
  agg_gemm_kernel<<<PD / 16, 256, 0, stream>>>(pl, LD, 0, lines, LD, tmp);
  lrl_gemm_kernel<<<PD / 16, 256, 0, stream>>>(tmp, points, 2.f, rel_W + 0 * HH,
      rel_b + 0 * HD, rel_g + 0 * HD, rel_be + 0 * HD, LPLb, 1);

  agg_gemm_kernel<<<LD / 16, 256, 0, stream>>>(pl, LD, 1, points, PD, tmp);
  lrl_gemm_kernel<<<LD / 16, 256, 0, stream>>>(tmp, lines, 2.f, rel_W + 1 * HH,
      rel_b + 1 * HD, rel_g + 1 * HD, rel_be + 1 * HD, PLPb, 1);

  agg_gemm_kernel<<<CD / 16, 256, 0, stream>>>(pc, CD, 1, points, PD, tmp);
  lrl_gemm_kernel<<<CD / 16, 256, 0, stream>>>(tmp, circles, 2.f, rel_W + 2 * HH,
      rel_b + 2 * HD, rel_g + 2 * HD, rel_be + 2 * HD, PCPb, 1);

  // 3) concat geo_t / geo_h  (device-to-device async copies: graph-capture safe)
  const size_t ROW = (size_t)HD * sizeof(float);
  hipMemcpyAsync(geo_t + (size_t)0 * HD,    points, PD * ROW, hipMemcpyDeviceToDevice, stream);
  hipMemcpyAsync(geo_t + (size_t)1024 * HD, lines,  LD * ROW, hipMemcpyDeviceToDevice, stream);
  hipMemcpyAsync(geo_t + (size_t)2048 * HD, LPLb,   PD * ROW, hipMemcpyDeviceToDevice, stream);
  hipMemcpyAsync(geo_t + (size_t)3072 * HD, PLPb,   LD * ROW, hipMemcpyDeviceToDevice, stream);
  hipMemcpyAsync(geo_t + (size_t)4096 * HD, PCPb,   CD * ROW, hipMemcpyDeviceToDevice, stream);
  hipMemcpyAsync(geo_t + (size_t)4224 * HD, head_s, NH * ROW, hipMemcpyDeviceToDevice, stream);
  hipMemcpyAsync(geo_h + (size_t)0 * HD,    points, PD * ROW, hipMemcpyDeviceToDevice, stream);
  hipMemcpyAsync(geo_h + (size_t)1024 * HD, LPLb,   PD * ROW, hipMemcpyDeviceToDevice, stream);
  hipMemcpyAsync(geo_h + (size_t)2048 * HD, PLPb,   LD * ROW, hipMemcpyDeviceToDevice, stream);
  hipMemcpyAsync(geo_h + (size_t)3072 * HD, PCPb,   CD * ROW, hipMemcpyDeviceToDevice, stream);
  mask_cat_kernel<<<(RT + 255) / 256, 256, 0, stream>>>(lpl_m, plp_m, pcp_m, mask_t, mask_h);

  // 4) gm(): two stacked LRL layers, applied in-place via tmp
  lrl_gemm_kernel<<<RT / 16, 256, 0, stream>>>(geo_t, nullptr, 1.f, gm_W + 0 * HH,
      gm_b + 0 * HD, gm_g + 0 * HD, gm_be + 0 * HD, tmp, 1);
  lrl_gemm_kernel<<<RT / 16, 256, 0, stream>>>(tmp, nullptr, 1.f, gm_W + 1 * HH,
      gm_b + 1 * HD, gm_g + 1 * HD, gm_be + 1 * HD, geo_t, 1);
  lrl_gemm_kernel<<<RH / 16, 256, 0, stream>>>(geo_h, nullptr, 1.f, gm_W + 0 * HH,
      gm_b + 0 * HD, gm_g + 0 * HD, gm_be + 0 * HD, tmp, 1);
  lrl_gemm_kernel<<<RH / 16, 256, 0, stream>>>(tmp, nullptr, 1.f, gm_W + 1 * HH,
      gm_b + 1 * HD, gm_g + 1 * HD, gm_be + 1 * HD, geo_h, 1);

  // 5) six score heads, factored: symW = sym@W1^T, geoW = geo@W1^T (raw GEMMs),
  //    then one wave32 per pair for ReLU/LN/dot/sigmoid.
  struct Head { const float* sym; int N; const float* geo; int R; const float* mask; };
  const Head heads[6] = {
      {text_s,  NT,  geo_t, RT, mask_t},
      {head_s,  NH,  geo_h, RH, mask_h},
      {angle_s, NA,  LPLb,  PD, lpl_m},
      {bar_s,   NB,  PLPb,  LD, plp_m},
      {para_s,  NPA, lines, LD, nullptr},
      {perp_s,  NPE, LPLb,  PD, lpl_m},
  };
  size_t ooff = 0;
  for (int h = 0; h < 6; ++h) {
    const Head& hd = heads[h];
    lrl_gemm_kernel<<<hd.N / 16, 256, 0, stream>>>(hd.sym, nullptr, 1.f,
        sc_W1 + (size_t)h * HH, nullptr, nullptr, nullptr, symWb, 0);
    lrl_gemm_kernel<<<hd.R / 16, 256, 0, stream>>>(hd.geo, nullptr, 1.f,
        sc_W1 + (size_t)h * HH, nullptr, nullptr, nullptr, geoWb, 0);
    int waves  = hd.N * hd.R;
    int blocks = (waves + 7) / 8;
    pair_kernel<<<blocks, 256, 0, stream>>>(symWb, geoWb, hd.mask,
        sc_b1 + (size_t)h * HD, sc_g + (size_t)h * HD, sc_be + (size_t)h * HD,
        sc_W2 + (size_t)h * HD, sc_b2 + h, out + ooff, hd.N, hd.R);
    ooff += (size_t)hd.N * hd.R;
  }
}